// BoundingBoxDiscipline_62457414419157
// MI455X (gfx1250) — compile-verified
//
#include <hip/hip_runtime.h>

// BoundingBoxDiscipline: bandwidth-bound masked bbox reduction.
// 512 MB read -> ~22 us floor at 23.3 TB/s. No matrix math -> WMMA is
// inapplicable; optimize the memory path (global_load_b128 streaming,
// global_prefetch_b8, wave32 packed-i16 butterfly with v_pk_min/max_i16).

#define SENT_MIN16 0x7fff   // i16 min-sentinel (y,x < 512 fit in i16)

namespace {
constexpr int kB = 16, kH = 512, kW = 512, kC = 16;
constexpr int kPix = kH * kW;                    // 262144 pixels / sample
constexpr int kBlk = 256;                        // 8 wave32 per block
constexpr int kBlocksPerSample = 64;
constexpr int kNT = kBlk * kBlocksPerSample;     // 16384 threads / sample (multiple of W)
constexpr int kIters = kPix / kNT;               // 16 pixels / thread
constexpr float kThreshPred = 0.3f;
constexpr float kThreshTrue = 0.5f;
}

__device__ __forceinline__ float hmax4(float4 v) {
  return fmaxf(fmaxf(v.x, v.y), fmaxf(v.z, v.w));  // lowers to v_max3_f32 chain
}

// CDNA5 VOP3P packed 2x i16 min/max (no portable intrinsic -> inline asm).
__device__ __forceinline__ int pk_min_i16(int a, int b) {
  int d;
  asm volatile("v_pk_min_i16 %0, %1, %2" : "=v"(d) : "v"(a), "v"(b));
  return d;
}
__device__ __forceinline__ int pk_max_i16(int a, int b) {
  int d;
  asm volatile("v_pk_max_i16 %0, %1, %2" : "=v"(d) : "v"(a), "v"(b));
  return d;
}

// ws layout (per sample b, 8 ints): pymin pxmin pymax pxmax tymin txmin tymax txmax
__global__ void bbox_init_kernel(int* __restrict__ ws) {
  int i = threadIdx.x;
  if (i < kB * 8) ws[i] = ((i & 3) < 2) ? 0x7fffffff : -1;  // mins=INT_MAX, maxes=-1
}

__global__ __launch_bounds__(kBlk) void bbox_reduce_kernel(
    const float* __restrict__ pred, const float* __restrict__ expd,
    int* __restrict__ ws) {
  const int b   = blockIdx.y;
  const int tid = blockIdx.x * kBlk + threadIdx.x;   // [0, kNT)
  const int x   = tid & (kW - 1);                    // constant column per thread
  int y         = tid >> 9;                          // row; advances by kNT/512
  const int ystep = kNT >> 9;

  const size_t sbase = (size_t)b * ((size_t)kPix * kC);
  const float4* __restrict__ p4 =
      reinterpret_cast<const float4*>(pred + sbase) + (size_t)tid * 4;
  const float4* __restrict__ e4 =
      reinterpret_cast<const float4*>(expd + sbase) + (size_t)tid * 4;

  int pymin = SENT_MIN16, pymax = -1, tymin = SENT_MIN16, tymax = -1;

#pragma unroll 2
  for (int k = 0; k < kIters; ++k) {
    const float4* pp = p4 + (size_t)k * ((size_t)kNT * 4);
    const float4* ee = e4 + (size_t)k * ((size_t)kNT * 4);
    if (k + 1 < kIters) {  // gfx1250 global_prefetch_b8, one 1MB stride ahead
      __builtin_prefetch(pp + (size_t)kNT * 4, 0, 0);
      __builtin_prefetch(ee + (size_t)kNT * 4, 0, 0);
    }
    float4 a0 = pp[0], a1 = pp[1], a2 = pp[2], a3 = pp[3];   // global_load_b128 x4
    float4 b0 = ee[0], b1 = ee[1], b2 = ee[2], b3 = ee[3];
    float mp = fmaxf(fmaxf(hmax4(a0), hmax4(a1)), fmaxf(hmax4(a2), hmax4(a3)));
    float me = fmaxf(fmaxf(hmax4(b0), hmax4(b1)), fmaxf(hmax4(b2), hmax4(b3)));
    if (mp > kThreshPred) { pymin = min(pymin, y); pymax = max(pymax, y); }
    if (me > kThreshTrue) { tymin = min(tymin, y); tymax = max(tymax, y); }
    y += ystep;
  }

  // x is constant per thread: contributes only if this thread found any pixel.
  int pxmin = (pymax >= 0) ? x : SENT_MIN16;
  int pxmax = (pymax >= 0) ? x : -1;
  int txmin = (tymax >= 0) ? x : SENT_MIN16;
  int txmax = (tymax >= 0) ? x : -1;

  // Pack the 8 values into 4 regs of 2x i16: lo = pred-mask, hi = true-mask.
  int ymins = (pymin & 0xffff) | (tymin << 16);
  int ymaxs = (pymax & 0xffff) | (tymax << 16);
  int xmins = (pxmin & 0xffff) | (txmin << 16);
  int xmaxs = (pxmax & 0xffff) | (txmax << 16);

  // wave32 butterfly with VOP3P packed i16 min/max (4 regs x 5 steps)
#pragma unroll
  for (int off = 16; off > 0; off >>= 1) {
    ymins = pk_min_i16(ymins, __shfl_xor(ymins, off, 32));
    ymaxs = pk_max_i16(ymaxs, __shfl_xor(ymaxs, off, 32));
    xmins = pk_min_i16(xmins, __shfl_xor(xmins, off, 32));
    xmaxs = pk_max_i16(xmaxs, __shfl_xor(xmaxs, off, 32));
  }

  __shared__ int s[8];
  if (threadIdx.x < 8) s[threadIdx.x] = ((threadIdx.x & 3) < 2) ? 0x7fffffff : -1;
  __syncthreads();
  if ((threadIdx.x & 31) == 0) {  // one lane per wave unpacks + 8 LDS atomics
    atomicMin(&s[0], (int)(short)(ymins & 0xffff));
    atomicMin(&s[1], (int)(short)(xmins & 0xffff));
    atomicMax(&s[2], (int)(short)(ymaxs & 0xffff));
    atomicMax(&s[3], (int)(short)(xmaxs & 0xffff));
    atomicMin(&s[4], ymins >> 16);
    atomicMin(&s[5], xmins >> 16);
    atomicMax(&s[6], ymaxs >> 16);
    atomicMax(&s[7], xmaxs >> 16);
  }
  __syncthreads();
  if (threadIdx.x < 8) {  // 8 global atomics per block
    int v = s[threadIdx.x];
    int* g = ws + b * 8 + threadIdx.x;
    if ((threadIdx.x & 3) < 2) atomicMin(g, v);
    else                       atomicMax(g, v);
  }
}

__global__ void bbox_finalize_kernel(const int* __restrict__ ws,
                                     float* __restrict__ out) {
  float v = 0.0f;
  int b = threadIdx.x;  // one wave32, lanes 0..15 carry data
  if (b < kB) {
    const int* s = ws + b * 8;
    int py1 = s[0], px1 = s[1], py2 = s[2], px2 = s[3];
    int ty1 = s[4], tx1 = s[5], ty2 = s[6], tx2 = s[7];
    if (py2 < 0) { py1 = 0; px1 = 0; py2 = 1; px2 = 1; }  // empty-mask fallback
    if (ty2 < 0) { ty1 = 0; tx1 = 0; ty2 = 1; tx2 = 1; }
    float parea = (float)((py2 - py1 + 1) * (px2 - px1 + 1));
    float tarea = (float)((ty2 - ty1 + 1) * (tx2 - tx1 + 1));
    float ap = fmaxf(parea - tarea, 0.0f) / (tarea + 1.0f);
    float pcy = 0.5f * (float)(py1 + py2), pcx = 0.5f * (float)(px1 + px2);
    float tcy = 0.5f * (float)(ty1 + ty2), tcx = 0.5f * (float)(tx1 + tx2);
    float dy = pcy - tcy, dx = pcx - tcx;
    float co = sqrtf(dy * dy + dx * dx) * (1.0f / 20.0f);
    v = ap + co;
  }
#pragma unroll
  for (int off = 8; off > 0; off >>= 1) v += __shfl_down(v, off, 32);
  if (threadIdx.x == 0) out[0] = 0.05f * (v * (1.0f / (float)kB));
}

extern "C" void kernel_launch(void* const* d_in, const int* in_sizes, int n_in,
                              void* d_out, int out_size, void* d_ws, size_t ws_size,
                              hipStream_t stream) {
  (void)in_sizes; (void)n_in; (void)out_size; (void)ws_size;
  const float* pred = (const float*)d_in[0];   // prediction_probs [16,512,512,16] f32
  const float* expd = (const float*)d_in[1];   // expected_onehot  [16,512,512,16] f32
  float* out = (float*)d_out;                  // scalar f32
  int* ws = (int*)d_ws;                        // 128 ints of bbox state

  hipLaunchKernelGGL(bbox_init_kernel, dim3(1), dim3(128), 0, stream, ws);
  hipLaunchKernelGGL(bbox_reduce_kernel, dim3(kBlocksPerSample, kB), dim3(kBlk),
                     0, stream, pred, expd, ws);
  hipLaunchKernelGGL(bbox_finalize_kernel, dim3(1), dim3(32), 0, stream, ws, out);
}